// SIFTNet_2267742733144
// MI455X (gfx1250) — compile-verified
//
#include <hip/hip_runtime.h>
#include <hip/hip_bf16.h>

typedef __attribute__((ext_vector_type(2))) float v2f;
typedef __attribute__((ext_vector_type(8))) float v8f;

#define IMG   1024
#define OUTD  1025
#define TILE  16
#define NT    65       // ceil(1025/16)
#define HREG  19       // hist region per tile (16 + 3 halo)
#define IREG  21       // input region per tile (19 + 2 sobel halo)

// band(k, j) = 1 if j <= k <= j+3  (4-wide ones band, pad offset 2)
__device__ __forceinline__ float bandv(int k, int j) {
    return ((unsigned)(k - j) <= 3u) ? 1.0f : 0.0f;
}

__global__ __launch_bounds__(256) void sift_fused(const float* __restrict__ x,
                                                  float* __restrict__ out) {
    __shared__ float  s_in[IREG][IREG + 3];   // input tile, zero-padded
    __shared__ float2 s_mi[HREG][21];         // {magnitude, bitcast(bin)}; col 19 = zero pad
    __shared__ float  s_T[8][20][17];         // per-wave horizontal-pass result; row 19 = zero pad

    const int tid = threadIdx.x;
    int t = blockIdx.x;
    const int b  = t / (NT * NT);
    t %= NT * NT;
    const int ty = t / NT, tx = t % NT;
    const int i0 = ty * TILE, j0 = tx * TILE;   // output-tile origin (0..1024)

    // ---------- stage input tile (rows i0-3..i0+17, cols j0-3..j0+17) ----------
    const float* xb = x + b * (IMG * IMG);
    for (int q = tid; q < IREG * IREG; q += 256) {
        int r = q / IREG, c = q % IREG;
        int gy = i0 - 3 + r, gx = j0 - 3 + c;
        float v = 0.0f;
        if ((unsigned)gy < (unsigned)IMG && (unsigned)gx < (unsigned)IMG)
            v = xb[gy * IMG + gx];
        s_in[r][c] = v;
    }
    // zero the K-padding column (x = 19) of the hist array
    if (tid < HREG) s_mi[tid][19] = make_float2(0.0f, 0.0f);
    __syncthreads();

    // ---------- Sobel + orientation argmax + magnitude (19x19) ----------
    for (int q = tid; q < HREG * HREG; q += 256) {
        int hy = q / HREG, hx = q % HREG;           // hist pixel, global (i0-2+hy, j0-2+hx)
        float p00 = s_in[hy    ][hx], p01 = s_in[hy    ][hx + 1], p02 = s_in[hy    ][hx + 2];
        float p10 = s_in[hy + 1][hx],                              p12 = s_in[hy + 1][hx + 2];
        float p20 = s_in[hy + 2][hx], p21 = s_in[hy + 2][hx + 1], p22 = s_in[hy + 2][hx + 2];
        float gxv = (p02 - p00) + 2.0f * (p12 - p10) + (p22 - p20);
        float gyv = (p20 - p00) + 2.0f * (p21 - p01) + (p22 - p02);
        float mag = sqrtf(gxv * gxv + gyv * gyv);
        // orientations at (0.125 + 0.25*o)*pi
        const float C0 = 0.92387953f, C1 = 0.38268343f;
        float cosv[8] = {  C0 * gxv + C1 * gyv,  C1 * gxv + C0 * gyv,
                          -C1 * gxv + C0 * gyv, -C0 * gxv + C1 * gyv,
                          -C0 * gxv - C1 * gyv, -C1 * gxv - C0 * gyv,
                           C1 * gxv - C0 * gyv,  C0 * gxv - C1 * gyv };
        int best = 0; float bv = cosv[0];
        #pragma unroll
        for (int o = 1; o < 8; ++o)
            if (cosv[o] > bv) { bv = cosv[o]; best = o; }   // first-max like jnp.argmax
        int gy2 = i0 - 2 + hy, gx2 = j0 - 2 + hx;
        bool ok = ((unsigned)gy2 < (unsigned)IMG) && ((unsigned)gx2 < (unsigned)IMG);
        s_mi[hy][hx] = make_float2(ok ? mag : 0.0f, __int_as_float(best));
    }
    __syncthreads();

    // ---------- per-wave: channel = wave id; 4x4 box filter as banded WMMA ----------
    const int wid  = tid >> 5;          // 0..7 -> orientation channel
    const int lane = tid & 31;
    const int n    = lane & 15;         // M for A operand / N for B,C,D operands
    const int half = lane >> 4;
    const int ch   = wid;

    // unconditional b64 load + cndmask (no EXEC branching)
    auto histAt = [&](int y, int xx) -> float {
        float2 p = s_mi[y][xx];                       // xx in 0..19 (19 = zero pad)
        return (__float_as_int(p.y) == ch) ? p.x : 0.0f;
    };

    // zero the K-padding row (y = 19) of this wave's T tile
    if (half == 0) s_T[wid][19][n] = 0.0f;

    // pass 1 (horizontal): T[y][j] = sum_x hist[y][x] * band(x, j), rows y=0..15 via WMMA
    v8f Tlo = {};
    #pragma unroll
    for (int s = 0; s < 5; ++s) {
        int k0 = 4 * s + 2 * half, k1 = k0 + 1;
        v2f a, bb;
        a.x  = histAt(n, k0);  a.y  = histAt(n, k1);   // A: hist (M=y, K=x)
        bb.x = bandv(k0, n);   bb.y = bandv(k1, n);    // B: band (K=x, N=j)
        Tlo = __builtin_amdgcn_wmma_f32_16x16x4_f32(
                  false, a, false, bb, (short)0, Tlo, false, false);
    }
    // scatter D layout (lane=N, vgpr=M/M+8) into LDS
    #pragma unroll
    for (int r = 0; r < 8; ++r)
        s_T[wid][r + 8 * half][n] = Tlo[r];
    // rows 16..18 of T: only 3 valid rows -> cheap VALU
    if (half == 0) {
        #pragma unroll
        for (int y = 16; y < HREG; ++y) {
            float acc = 0.0f;
            #pragma unroll
            for (int d = 0; d < 4; ++d)
                acc += histAt(y, n + d);
            s_T[wid][y][n] = acc;
        }
    }
    // (same-wave LDS RAW: compiler inserts s_wait_dscnt; no barrier needed)

    // pass 2 (vertical): Out[i][j] = sum_y band(y, i) * T[y][j] via WMMA
    v8f Out = {};
    #pragma unroll
    for (int s = 0; s < 5; ++s) {
        int k0 = 4 * s + 2 * half, k1 = k0 + 1;
        v2f a, bb;
        a.x  = bandv(k0, n);                          // A: band (M=i, K=y)
        a.y  = bandv(k1, n);
        bb.x = s_T[wid][k0][n];                       // B: T (K=y, N=j); row 19 = zero pad
        bb.y = s_T[wid][k1][n];
        Out = __builtin_amdgcn_wmma_f32_16x16x4_f32(
                  false, a, false, bb, (short)0, Out, false, false);
    }

    // ---------- store output tile (8, 8, 1025, 1025) ----------
    float* ob = out + (size_t)(b * 8 + ch) * (OUTD * OUTD);
    #pragma unroll
    for (int r = 0; r < 8; ++r) {
        int oi = i0 + r + 8 * half;                   // D: vgpr r -> row r / r+8
        int oj = j0 + n;
        if (oi < OUTD && oj < OUTD)
            ob[oi * OUTD + oj] = Out[r];
    }
}

extern "C" void kernel_launch(void* const* d_in, const int* in_sizes, int n_in,
                              void* d_out, int out_size, void* d_ws, size_t ws_size,
                              hipStream_t stream) {
    const float* x = (const float*)d_in[0];   // (8,1,1024,1024) f32
    float* out = (float*)d_out;               // (8,8,1025,1025) f32
    dim3 grid(NT * NT * 8), block(256);
    hipLaunchKernelGGL(sift_fused, grid, block, 0, stream, x, out);
}